// HiggsAudioV2TokenizerVectorQuantization_45406394253484
// MI455X (gfx1250) — compile-verified
//
#include <hip/hip_runtime.h>

// ---------------------------------------------------------------------------
// HiggsAudio VQ: encode-project / euclidean-codebook-argmax / decode-project
// fp32 on CDNA5 matrix pipe (V_WMMA_F32_16X16X4_F32, wave32) with
// GLOBAL_LOAD_ASYNC_TO_LDS_B128 double-buffered staging (ASYNCcnt).
// ---------------------------------------------------------------------------

typedef __attribute__((ext_vector_type(2))) float v2f;
typedef __attribute__((ext_vector_type(8))) float v8f;

__device__ __forceinline__ v8f wmma_f32x4(v2f a, v2f b, v8f c) {
  return __builtin_amdgcn_wmma_f32_16x16x4_f32(
      /*neg_a=*/false, a, /*neg_b=*/false, b,
      /*c_mod=*/(short)0, c, /*reuse_a=*/false, /*reuse_b=*/false);
}

__device__ __forceinline__ v8f zero8() {
  v8f z = {0.f, 0.f, 0.f, 0.f, 0.f, 0.f, 0.f, 0.f};
  return z;
}

// Async 16B global -> LDS copy (per-lane), tracked by ASYNCcnt.
// LDS offset = low 32 bits of the generic pointer (wave-relative).
__device__ __forceinline__ void async_b128(unsigned lds_off, const float* gsrc) {
  asm volatile("global_load_async_to_lds_b128 %0, %1, off"
               :: "v"(lds_off), "v"((unsigned long long)(const void*)gsrc)
               : "memory");
}
#define WAIT_ASYNC0() asm volatile("s_wait_asynccnt 0x0" ::: "memory")

__device__ __forceinline__ unsigned lds_off(const void* p) {
  return (unsigned)(unsigned long long)p;
}

constexpr int Bn = 8, Hn = 1024, Tn = 8192, Dn = 64, Kn = 1024;

// ---------------------------------------------------------------------------
// Kernel 0: codebook squared norms  e2[k] = sum_d embed[k][d]^2
// ---------------------------------------------------------------------------
__global__ void __launch_bounds__(256) vq_e2_kernel(const float* __restrict__ embed,
                                                    float* __restrict__ e2) {
  int k = blockIdx.x * 256 + threadIdx.x;
  if (k < Kn) {
    const float4* row = (const float4*)(embed + (size_t)k * Dn);
    float s = 0.f;
#pragma unroll
    for (int i = 0; i < Dn / 4; ++i) {
      float4 v = row[i];
      s += v.x * v.x + v.y * v.y + v.z * v.z + v.w * v.w;
    }
    e2[k] = s;
  }
}

// ---------------------------------------------------------------------------
// Kernel 1: X[b][d][t] = sum_h w_in[d][h] * hidden[b][h][t] + b_in[d]
// WG (256 thr, 8 waves): 64 (all D) x 256 (t) tile; K=1024 in 64-row chunks,
// double-buffered via async-to-LDS. hidden tile [64][272] row-major (B-frags:
// two b32, halves 32 banks apart); w_in chunk [64 m][68 k] row-major (A-frags:
// one b64, conflict-free).
// ---------------------------------------------------------------------------
#define K1_TS 272      // hidden tile row stride (floats): 2*272 mod 64 == 32
#define K1_WS 68       // w chunk row stride (floats): bank stride 4, b64-clean
#define K1_TILE_F (64 * K1_TS)
#define K1_W_F (64 * K1_WS)
__global__ void __launch_bounds__(256) vq_proj_in_kernel(
    const float* __restrict__ hidden, const float* __restrict__ w_in,
    const float* __restrict__ b_in, float* __restrict__ Xd) {
  extern __shared__ float smem1[];

  const int tid   = threadIdx.x;
  const int b     = blockIdx.y;
  const int t0    = blockIdx.x * 256;
  const int lane  = tid & 31;
  const int wave  = tid >> 5;
  const int lhalf = lane >> 4;
  const int lmod  = lane & 15;
  const int wn0   = wave * 32;

  const int r   = tid >> 2;  // hidden tile row 0..63 / w row (m) 0..63
  const int seg = tid & 3;   // float4 segment

  v8f acc[4][2];
#pragma unroll
  for (int mt = 0; mt < 4; ++mt)
#pragma unroll
    for (int nt = 0; nt < 2; ++nt) acc[mt][nt] = zero8();

  // async-stage chunk kc into buffer bi: 16 (hidden) + 4 (w) b128 per thread
  auto issue_chunk = [&](int kc, int bi) {
    float* tb_base = smem1 + bi * K1_TILE_F;
    float* wb_base = smem1 + 2 * K1_TILE_F + bi * K1_W_F;
    const float* hrow = hidden + ((size_t)b * Hn + (size_t)kc * 64 + r) * Tn + t0;
    const unsigned tb = lds_off(tb_base + r * K1_TS);
#pragma unroll
    for (int i = 0; i < 16; ++i) {
      const int c4 = seg + i * 4;
      async_b128(tb + c4 * 16, hrow + c4 * 4);
    }
    const float* wrow = w_in + (size_t)r * Hn + kc * 64;
    const unsigned wb = lds_off(wb_base + r * K1_WS);
#pragma unroll
    for (int i = 0; i < 4; ++i) {
      const int c4 = seg * 4 + i;
      async_b128(wb + c4 * 16, wrow + c4 * 4);
    }
  };

  issue_chunk(0, 0);
  for (int kc = 0; kc < Hn / 64; ++kc) {
    const int cur = kc & 1;
    WAIT_ASYNC0();          // this wave's pending asyncs (chunk kc) complete
    __syncthreads();        // all waves' chunk kc visible; prev buffer free
    if (kc < Hn / 64 - 1) issue_chunk(kc + 1, cur ^ 1);  // overlaps compute
    const float* tl = smem1 + cur * K1_TILE_F;
    const float* wl = smem1 + 2 * K1_TILE_F + cur * K1_W_F;
#pragma unroll 4
    for (int ks = 0; ks < 16; ++ks) {
      const int k0 = ks * 4 + 2 * lhalf;  // this half's K rows: k0, k0+1
      v2f afrag[4];
#pragma unroll
      for (int mt = 0; mt < 4; ++mt)
        afrag[mt] = *(const v2f*)&wl[(mt * 16 + lmod) * K1_WS + k0];
      v2f bfrag[2];
#pragma unroll
      for (int nt = 0; nt < 2; ++nt) {
        bfrag[nt].x = tl[(k0 + 0) * K1_TS + wn0 + nt * 16 + lmod];
        bfrag[nt].y = tl[(k0 + 1) * K1_TS + wn0 + nt * 16 + lmod];
      }
#pragma unroll
      for (int mt = 0; mt < 4; ++mt)
#pragma unroll
        for (int nt = 0; nt < 2; ++nt)
          acc[mt][nt] = wmma_f32x4(afrag[mt], bfrag[nt], acc[mt][nt]);
    }
  }
  // epilogue: + b_in, store d-major (64B per half-wave)
#pragma unroll
  for (int mt = 0; mt < 4; ++mt) {
#pragma unroll
    for (int i = 0; i < 8; ++i) {
      const int d = mt * 16 + i + 8 * lhalf;
      const float bias = b_in[d];
#pragma unroll
      for (int nt = 0; nt < 2; ++nt) {
        const int t = t0 + wn0 + nt * 16 + lmod;
        Xd[((size_t)b * Dn + d) * Tn + t] = acc[mt][nt][i] + bias;
      }
    }
  }
}

// ---------------------------------------------------------------------------
// Kernel 2: S^T[k][t] = embed[k][:] . X[:, t]; argmax_k (2*S - e2[k]).
// Whole codebook async-copied row-major into LDS [1024][68] (278 KB < 320 KB):
// A-frags are single conflict-free ds_b64. Each wave owns 16 t's; X fragments
// resident in 32 VGPRs; 64 code-tiles x 16 WMMA; exact first-index tie-break;
// cross-half combine via __shfl_xor(16) (wave32).
// ---------------------------------------------------------------------------
#define K2_ES 68
__global__ void __launch_bounds__(256) vq_argmax_kernel(
    const float* __restrict__ Xd, const float* __restrict__ embed,
    const float* __restrict__ e2g, int* __restrict__ ind) {
  extern __shared__ float smem2[];
  float* eL  = smem2;                  // [1024][68]
  float* e2s = smem2 + Kn * K2_ES;     // [1024]

  const int tid   = threadIdx.x;
  const int b     = blockIdx.y;
  const int lane  = tid & 31;
  const int wave  = tid >> 5;
  const int lhalf = lane >> 4;
  const int lmod  = lane & 15;

  // async-stage codebook (row-major, padded stride) + norms
#pragma unroll
  for (int kb = 0; kb < 4; ++kb) {
    const int k = kb * 256 + tid;
    const float* grow = embed + (size_t)k * Dn;
    const unsigned lb = lds_off(&eL[k * K2_ES]);
#pragma unroll
    for (int i = 0; i < 16; ++i) async_b128(lb + i * 16, grow + i * 4);
    e2s[k] = e2g[k];
  }
  WAIT_ASYNC0();
  __syncthreads();

  const int t0 = blockIdx.x * 128 + wave * 16;
  v2f bfrag[16];  // resident: X[d][t0+lmod], d = 4s + 2*lhalf (+1)
#pragma unroll
  for (int s = 0; s < 16; ++s) {
    const int d = s * 4 + 2 * lhalf;
    bfrag[s].x = Xd[((size_t)b * Dn + d + 0) * Tn + t0 + lmod];
    bfrag[s].y = Xd[((size_t)b * Dn + d + 1) * Tn + t0 + lmod];
  }

  float bestv = -3.402823466e38f;
  int   besti = 0;
  for (int kt = 0; kt < Kn / 16; ++kt) {
    v8f c = zero8();
#pragma unroll
    for (int s = 0; s < 16; ++s) {
      v2f a = *(const v2f*)&eL[(kt * 16 + lmod) * K2_ES + s * 4 + 2 * lhalf];
      c = wmma_f32x4(a, bfrag[s], c);
    }
#pragma unroll
    for (int i = 0; i < 8; ++i) {
      const int code = kt * 16 + i + 8 * lhalf;
      const float score = 2.0f * c[i] - e2s[code];
      if (score > bestv) { bestv = score; besti = code; }  // strict > keeps first
    }
  }
  const float ov = __shfl_xor(bestv, 16, 32);
  const int   oi = __shfl_xor(besti, 16, 32);
  if (ov > bestv || (ov == bestv && oi < besti)) { bestv = ov; besti = oi; }
  if (lhalf == 0) ind[(size_t)b * Tn + t0 + lmod] = besti;
}

// ---------------------------------------------------------------------------
// Kernel 3: out[b][h][t] = sum_d w_out[h][d] * embed[ind[b][t]][d] + b_out[h]
// WG: 256 t's x 128 h's (8 waves x one 16-row m-tile); grid.z covers H/128.
// Quantized rows gathered t-major into LDS [256][68] with b128 stores:
// B-frags are single conflict-free ds_b64; w_out fragments register-resident.
// ---------------------------------------------------------------------------
#define K3_QS 68
__global__ void __launch_bounds__(256) vq_proj_out_kernel(
    const int* __restrict__ ind, const float* __restrict__ embed,
    const float* __restrict__ w_out, const float* __restrict__ b_out,
    float* __restrict__ out) {
  extern __shared__ float smem3[];
  float* Q = smem3;  // [256 t][68 d]

  const int tid   = threadIdx.x;
  const int b     = blockIdx.y;
  const int t0    = blockIdx.x * 256;
  const int lane  = tid & 31;
  const int wave  = tid >> 5;
  const int lhalf = lane >> 4;
  const int lmod  = lane & 15;

  {  // gather: row tid = codebook row ind[t0+tid] (16 x ds_store_b128)
    const int code = ind[(size_t)b * Tn + t0 + tid];
    const float4* row = (const float4*)(embed + (size_t)code * Dn);
    float4* qrow = (float4*)&Q[tid * K3_QS];
#pragma unroll
    for (int i = 0; i < 16; ++i) qrow[i] = row[i];
  }
  __syncthreads();

  const int m0 = (blockIdx.z * 8 + wave) * 16;
  v2f afrag[16];  // resident: w_out[m0+lmod][d]
#pragma unroll
  for (int s = 0; s < 16; ++s) {
    const int d = s * 4 + 2 * lhalf;
    afrag[s] = *(const v2f*)(w_out + (size_t)(m0 + lmod) * Dn + d);
  }
  float bias[8];
#pragma unroll
  for (int i = 0; i < 8; ++i) bias[i] = b_out[m0 + i + 8 * lhalf];

  for (int tt = 0; tt < 16; ++tt) {
    v8f c = zero8();
#pragma unroll
    for (int s = 0; s < 16; ++s) {
      v2f bf = *(const v2f*)&Q[(tt * 16 + lmod) * K3_QS + s * 4 + 2 * lhalf];
      c = wmma_f32x4(afrag[s], bf, c);
    }
#pragma unroll
    for (int i = 0; i < 8; ++i) {
      const int h = m0 + i + 8 * lhalf;
      out[((size_t)b * Hn + h) * Tn + t0 + tt * 16 + lmod] = c[i] + bias[i];
    }
  }
}

// ---------------------------------------------------------------------------
extern "C" void kernel_launch(void* const* d_in, const int* in_sizes, int n_in,
                              void* d_out, int out_size, void* d_ws, size_t ws_size,
                              hipStream_t stream) {
  const float* hidden = (const float*)d_in[0];  // [B,H,T]
  const float* w_in   = (const float*)d_in[1];  // [D,H]
  const float* b_in   = (const float*)d_in[2];  // [D]
  const float* embed  = (const float*)d_in[3];  // [K,D]
  const float* w_out  = (const float*)d_in[4];  // [H,D]
  const float* b_out  = (const float*)d_in[5];  // [H]
  float* out = (float*)d_out;                   // [B,H,T]

  float* Xd = (float*)d_ws;                     // B*D*T floats (16 MB)
  float* e2 = Xd + (size_t)Bn * Dn * Tn;        // K floats
  int* indb = (int*)(e2 + Kn);                  // B*T ints

  const size_t lds1 = (size_t)(2 * K1_TILE_F + 2 * K1_W_F) * sizeof(float); // 174080
  const size_t lds2 = (size_t)(Kn * K2_ES + Kn) * sizeof(float);            // 282624
  const size_t lds3 = (size_t)(256 * K3_QS) * sizeof(float);                // 69632

  static bool attr_done = false;  // idempotent attribute setup (not a work guard)
  if (!attr_done) {
    (void)hipFuncSetAttribute((const void*)vq_proj_in_kernel,
                              hipFuncAttributeMaxDynamicSharedMemorySize, (int)lds1);
    (void)hipFuncSetAttribute((const void*)vq_argmax_kernel,
                              hipFuncAttributeMaxDynamicSharedMemorySize, (int)lds2);
    (void)hipFuncSetAttribute((const void*)vq_proj_out_kernel,
                              hipFuncAttributeMaxDynamicSharedMemorySize, (int)lds3);
    attr_done = true;
  }

  vq_e2_kernel<<<dim3(Kn / 256), 256, 0, stream>>>(embed, e2);
  vq_proj_in_kernel<<<dim3(Tn / 256, Bn), 256, lds1, stream>>>(hidden, w_in, b_in, Xd);
  vq_argmax_kernel<<<dim3(Tn / 128, Bn), 256, lds2, stream>>>(Xd, embed, e2, indb);
  vq_proj_out_kernel<<<dim3(Tn / 256, Bn, Hn / 128), 256, lds3, stream>>>(
      indb, embed, w_out, b_out, out);
}